// EdgeDotGATConv_84567906058978
// MI455X (gfx1250) — compile-verified
//
#include <hip/hip_runtime.h>
#include <hip/hip_bf16.h>

// Problem constants (match reference)
#define NN 20000
#define EE 320000
#define HH 4
#define FF 64
#define CC 256   // H*F = D_N = D_E

typedef __attribute__((ext_vector_type(2))) float v2f;
typedef __attribute__((ext_vector_type(4))) float v4f;
typedef __attribute__((ext_vector_type(8))) float v8f;

// ---------------------------------------------------------------------------
// Pack W (256x256 row-major) into K-pair interleaved layout so each WMMA B
// fragment is one contiguous b64 per lane:
//   Wp2[(k>>1)*256 + col] = { W[k][col], W[k+1][col] }   (k even)
// ---------------------------------------------------------------------------
__global__ __launch_bounds__(256) void gat_pack_w(const float* __restrict__ W,
                                                  float* __restrict__ Wp) {
  const int tid = blockIdx.x * 256 + threadIdx.x;  // 128 k-pairs * 256 cols
  const int col = tid & (CC - 1);
  const int kp  = tid >> 8;
  v2f p;
  p.x = W[(size_t)(2 * kp) * CC + col];
  p.y = W[(size_t)(2 * kp + 1) * CC + col];
  *(v2f*)(Wp + (size_t)tid * 2) = p;
}

// ---------------------------------------------------------------------------
// WMMA mainloop: one wave computes a 32(M) x 128(N) fp32 tile of A[Mx256] @ W
// (two 16-row blocks sharing B fragments). Per K-step: 2 A b64 loads +
// 8 B b64 loads + 16 v_wmma_f32_16x16x4_f32.
// ---------------------------------------------------------------------------
__device__ __forceinline__ void wmma_tile_32x128(const float* __restrict__ A,
                                                 const float* __restrict__ Wp,
                                                 int row0, int colbase, int M,
                                                 v8f acc0[8], v8f acc1[8]) {
  const int lane = threadIdx.x & 31;
  const int half = lane >> 4;       // which 16-lane half
  const int lm   = lane & 15;
  int ar0 = row0 + lm;       if (ar0 > M - 1) ar0 = M - 1;
  int ar1 = row0 + 16 + lm;  if (ar1 > M - 1) ar1 = M - 1;
  const float* aptr0 = A + (size_t)ar0 * CC + 2 * half;
  const float* aptr1 = A + (size_t)ar1 * CC + 2 * half;

  const v8f vzero = {0.f, 0.f, 0.f, 0.f, 0.f, 0.f, 0.f, 0.f};
#pragma unroll
  for (int t = 0; t < 8; ++t) { acc0[t] = vzero; acc1[t] = vzero; }

  for (int k0 = 0; k0 < CC; k0 += 4) {
    // A fragments: lane<16 holds {K=k0,k0+1}, lane>=16 holds {K=k0+2,k0+3}
    v2f a0 = *(const v2f*)(aptr0 + k0);
    v2f a1 = *(const v2f*)(aptr1 + k0);
    // Packed B: rows (k0+2*half, k0+2*half+1) at col -> one b64
    const float* wrow = Wp + (size_t)((k0 >> 1) + half) * (2 * CC) +
                        2 * (colbase + lm);
    v2f bf[8];
#pragma unroll
    for (int t = 0; t < 8; ++t) bf[t] = *(const v2f*)(wrow + t * 32);
#pragma unroll
    for (int t = 0; t < 8; ++t) {
      acc0[t] = __builtin_amdgcn_wmma_f32_16x16x4_f32(
          false, a0, false, bf[t], (short)0, acc0[t], false, false);
      acc1[t] = __builtin_amdgcn_wmma_f32_16x16x4_f32(
          false, a1, false, bf[t], (short)0, acc1[t], false, false);
    }
  }
}

// ---------------------------------------------------------------------------
// init: zero agg output, zero softmax denominators, set segment-max to -inf
// ---------------------------------------------------------------------------
__global__ __launch_bounds__(256) void gat_init(float* __restrict__ out,
                                                float* __restrict__ mmax,
                                                float* __restrict__ s) {
  const int tid = blockIdx.x * 256 + threadIdx.x;  // grid covers N*256
  out[tid] = 0.f;
  if (tid < NN * HH) {
    mmax[tid] = -3.0e38f;
    s[tid] = 0.f;
  }
}

// ---------------------------------------------------------------------------
// ft = nfeat @ W_node   (N x 256)
// ---------------------------------------------------------------------------
__global__ __launch_bounds__(256) void gat_node_gemm(const float* __restrict__ A,
                                                     const float* __restrict__ Wp,
                                                     float* __restrict__ ft, int M) {
  const int wave    = threadIdx.x >> 5;
  const int row0    = blockIdx.x * 128 + (wave >> 1) * 32;
  const int colbase = (wave & 1) * 128;
  v8f acc0[8], acc1[8];
  wmma_tile_32x128(A, Wp, row0, colbase, M, acc0, acc1);

  const int lane = threadIdx.x & 31, half = lane >> 4, lm = lane & 15;
#pragma unroll
  for (int r = 0; r < 8; ++r) {
    const int rowa = row0 + r + 8 * half;       // C layout: VGPR r, half -> M
    const int rowb = rowa + 16;
    if (rowa < M) {
#pragma unroll
      for (int t = 0; t < 8; ++t)
        ft[(size_t)rowa * CC + colbase + t * 16 + lm] = acc0[t][r];
    }
    if (rowb < M) {
#pragma unroll
      for (int t = 0; t < 8; ++t)
        ft[(size_t)rowb * CC + colbase + t * 16 + lm] = acc1[t][r];
    }
  }
}

// ---------------------------------------------------------------------------
// Fused edge stage:
//   eft       = efeat @ W_edge           (WMMA mainloop)
//   ft_prime  = eft + ft[src]            (gather-add in epilogue)
//   a[e][h]   = <ft_prime, ft[dst]>      (per-head dot, shfl_xor reduction)
//   m[d][h]   = segment-max via global_atomic_max_num_f32
// Each wave owns cols [colbase, colbase+128) = exactly 2 heads -> each (e,h)
// is produced by exactly one wave, so a[] needs no atomics.
// ---------------------------------------------------------------------------
__device__ __forceinline__ void edge_epilogue_rowblk(
    const v8f acc[8], int e, int colbase, int h0, int lm,
    const int* __restrict__ src, const int* __restrict__ dst,
    const float* __restrict__ ft, float* __restrict__ ftp,
    float* __restrict__ a, float* __restrict__ mmax, int r) {
  const int se = src[e];
  const int de = dst[e];
  const float* fs = ft  + (size_t)se * CC + colbase + lm;
  const float* fd = ft  + (size_t)de * CC + colbase + lm;
  float*       fo = ftp + (size_t)e  * CC + colbase + lm;
  float ps0 = 0.f, ps1 = 0.f;
#pragma unroll
  for (int t = 0; t < 8; ++t) {
    float v = acc[t][r] + fs[t * 16];   // u_add_e
    fo[t * 16] = v;                     // persist ft_prime
    float p = v * fd[t * 16];           // e_dot_v partial
    if (t < 4) ps0 += p; else ps1 += p;
  }
#pragma unroll
  for (int msk = 8; msk; msk >>= 1) {   // reduce within the 16-lane half
    ps0 += __shfl_xor(ps0, msk, 32);
    ps1 += __shfl_xor(ps1, msk, 32);
  }
  if (lm == 0) {
    a[e * HH + h0]     = ps0;
    a[e * HH + h0 + 1] = ps1;
    unsafeAtomicMax(&mmax[de * HH + h0],     ps0);  // global_atomic_max_num_f32
    unsafeAtomicMax(&mmax[de * HH + h0 + 1], ps1);
  }
}

__global__ __launch_bounds__(256) void gat_edge_gemm_fused(
    const float* __restrict__ efeat, const float* __restrict__ Wp,
    const int* __restrict__ src, const int* __restrict__ dst,
    const float* __restrict__ ft, float* __restrict__ ftp,
    float* __restrict__ a, float* __restrict__ mmax) {
  const int wave    = threadIdx.x >> 5;
  const int e0      = blockIdx.x * 128 + (wave >> 1) * 32;
  const int colbase = (wave & 1) * 128;
  v8f acc0[8], acc1[8];
  wmma_tile_32x128(efeat, Wp, e0, colbase, EE, acc0, acc1);

  const int lane = threadIdx.x & 31, half = lane >> 4, lm = lane & 15;
  const int h0 = colbase >> 6;  // first of the two heads this wave owns

#pragma unroll
  for (int r = 0; r < 8; ++r) {
    edge_epilogue_rowblk(acc0, e0 + r + 8 * half,      colbase, h0, lm,
                         src, dst, ft, ftp, a, mmax, r);
    edge_epilogue_rowblk(acc1, e0 + 16 + r + 8 * half, colbase, h0, lm,
                         src, dst, ft, ftp, a, mmax, r);
  }
}

// ---------------------------------------------------------------------------
// ex = exp(a - m[dst]); s[dst] += ex   (a overwritten with ex)
// ---------------------------------------------------------------------------
__global__ __launch_bounds__(256) void gat_exp_sum(const int* __restrict__ dst,
                                                   const float* __restrict__ mmax,
                                                   float* __restrict__ a,
                                                   float* __restrict__ s) {
  const int tid = blockIdx.x * 256 + threadIdx.x;  // E*H threads
  const int e = tid >> 2, h = tid & 3;
  const int de = dst[e];
  const float v = __expf(a[tid] - mmax[de * HH + h]);
  a[tid] = v;
  unsafeAtomicAdd(&s[de * HH + h], v);             // global_atomic_add_f32
}

// ---------------------------------------------------------------------------
// agg[dst] += (ex / s[dst] / sqrt(F)) * ft_prime  — one wave per edge,
// lane handles 8 contiguous cols (one head per 8-lane group), f32 atomics.
// ---------------------------------------------------------------------------
__global__ __launch_bounds__(256) void gat_aggregate(const int* __restrict__ dst,
                                                     const float* __restrict__ ex,
                                                     const float* __restrict__ s,
                                                     const float* __restrict__ ftp,
                                                     float* __restrict__ out) {
  const int wave = threadIdx.x >> 5;
  const int lane = threadIdx.x & 31;
  const int e    = blockIdx.x * 8 + wave;
  const int de   = dst[e];
  const int head = lane >> 3;  // 64-col heads, 8 cols per lane
  const float sa = ex[e * HH + head] / s[de * HH + head] * 0.125f;  // 1/sqrt(64)

  const float* fp = ftp + (size_t)e * CC + lane * 8;
  float*       op = out + (size_t)de * CC + lane * 8;
  v4f x0 = *(const v4f*)fp;
  v4f x1 = *(const v4f*)(fp + 4);
#pragma unroll
  for (int j = 0; j < 4; ++j) unsafeAtomicAdd(op + j, sa * x0[j]);
#pragma unroll
  for (int j = 0; j < 4; ++j) unsafeAtomicAdd(op + 4 + j, sa * x1[j]);
}

// ---------------------------------------------------------------------------
extern "C" void kernel_launch(void* const* d_in, const int* in_sizes, int n_in,
                              void* d_out, int out_size, void* d_ws, size_t ws_size,
                              hipStream_t stream) {
  const float* nfeat = (const float*)d_in[0];
  const float* efeat = (const float*)d_in[1];
  const int*   src   = (const int*)d_in[2];
  const int*   dst   = (const int*)d_in[3];
  const float* Wn    = (const float*)d_in[4];
  const float* We    = (const float*)d_in[5];
  float* out = (float*)d_out;

  // workspace layout (floats): ft | ft_prime | a/ex | m | s | Wn_packed | We_packed
  float* ft  = (float*)d_ws;
  float* ftp = ft  + (size_t)NN * CC;
  float* a   = ftp + (size_t)EE * CC;
  float* m   = a   + (size_t)EE * HH;
  float* s   = m   + (size_t)NN * HH;
  float* wnp = s   + (size_t)NN * HH;
  float* wep = wnp + (size_t)CC * CC;

  gat_pack_w<<<128, 256, 0, stream>>>(Wn, wnp);
  gat_pack_w<<<128, 256, 0, stream>>>(We, wep);
  gat_init<<<NN, 256, 0, stream>>>(out, m, s);                       // N*256 threads
  gat_node_gemm<<<(NN + 127) / 128, 256, 0, stream>>>(nfeat, wnp, ft, NN);
  gat_edge_gemm_fused<<<EE / 128, 256, 0, stream>>>(efeat, wep, src, dst, ft, ftp, a, m);
  gat_exp_sum<<<(EE * HH) / 256, 256, 0, stream>>>(dst, m, a, s);
  gat_aggregate<<<EE / 8, 256, 0, stream>>>(dst, a, s, ftp, out);
}